// ClassifierHead_40269613367577
// MI455X (gfx1250) — compile-verified
//
#include <hip/hip_runtime.h>
#include <hip/hip_bf16.h>

#define N_NODES   100000
#define N_DIM     512
#define N_REL     64
#define N_TRIPLES 200000
#define TILE_M    16
#define NCOLS     128   // 64 s-score cols | 64 o-score cols

typedef __attribute__((ext_vector_type(2))) float v2f;
typedef __attribute__((ext_vector_type(8))) float v8f;

// ---------------------------------------------------------------------------
// Phase 1: P[node, c] = dot(E[node], W[c%64][ (c<64?0:512) + : ])
// One block = one 16-node tile. 8 waves, each owns a 16-column tile.
// WMMA f32 16x16x4, K swept 0..511 in steps of 4.
// ---------------------------------------------------------------------------
__global__ __launch_bounds__(256) void proj_kernel(const float* __restrict__ E,
                                                   const float* __restrict__ W,
                                                   float* __restrict__ P) {
    __shared__ float ldsE[TILE_M * N_DIM];            // 32 KB E tile
    const int tid      = threadIdx.x;
    const int nodeBase = blockIdx.x * TILE_M;

    // Cooperative, coalesced float4 stage of the 16x512 f32 tile (2048 float4s)
    const float4* Eg = reinterpret_cast<const float4*>(E + (size_t)nodeBase * N_DIM);
    float4*       El = reinterpret_cast<float4*>(ldsE);
#pragma unroll
    for (int i = 0; i < 8; ++i) {
        El[i * 256 + tid] = Eg[i * 256 + tid];
    }
    __syncthreads();

    const int wave  = tid >> 5;            // 0..7  -> column tile
    const int lane  = tid & 31;
    const int n16   = lane & 15;           // M (A) / N (B,D) index within tile
    const int khalf = (lane >> 4) << 1;    // 0 for lanes 0-15, 2 for lanes 16-31

    const int c0    = wave * 16;           // output column base
    const int rBase = c0 & 63;             // relation base for this column tile
    const int wOff  = (c0 < N_REL) ? 0 : N_DIM;   // Ws half vs Wo half of W row

    // A fragment source (LDS): lane holds E[nodeBase+n16][k+khalf], [k+khalf+1]
    const float* aPtr = ldsE + n16 * N_DIM + khalf;
    // B fragment source (global, L2-resident 256 KB):
    // lane holds W[rBase+n16][wOff + k + khalf], [ .. +1]
    const float* bPtr = W + (size_t)(rBase + n16) * (2 * N_DIM) + wOff + khalf;

    v8f acc = {0.f, 0.f, 0.f, 0.f, 0.f, 0.f, 0.f, 0.f};
#pragma unroll 4
    for (int k = 0; k < N_DIM; k += 4) {
        v2f a = *reinterpret_cast<const v2f*>(aPtr + k);   // ds_load_b64
        v2f b = *reinterpret_cast<const v2f*>(bPtr + k);   // global_load_b64
        // (neg_a, A, neg_b, B, c_mod, C, reuse_a, reuse_b)
        acc = __builtin_amdgcn_wmma_f32_16x16x4_f32(false, a, false, b,
                                                    (short)0, acc, false, false);
    }

    // D layout: VGPR j -> M=j (lanes 0-15) / M=8+j (lanes 16-31), N=n16
    const int rowOff = (lane >> 4) * 8;
    float* outBase = P + (size_t)nodeBase * NCOLS + c0 + n16;
#pragma unroll
    for (int j = 0; j < 8; ++j) {
        outBase[(size_t)(rowOff + j) * NCOLS] = acc[j];
    }
}

// ---------------------------------------------------------------------------
// Phase 2: score[t] = P[s,r] + P[o,64+r] + b[r]   (4-byte gathers, L2-resident)
// ---------------------------------------------------------------------------
__global__ void score_kernel(const int* __restrict__ tri,
                             const float* __restrict__ P,
                             const float* __restrict__ b,
                             float* __restrict__ out) {
    int t = blockIdx.x * blockDim.x + threadIdx.x;
    if (t >= N_TRIPLES) return;
    int s = tri[t];
    int r = tri[N_TRIPLES + t];
    int o = tri[2 * N_TRIPLES + t];
    out[t] = P[(size_t)s * NCOLS + r] + P[(size_t)o * NCOLS + N_REL + r] + b[r];
}

// ---------------------------------------------------------------------------
// Fallback (workspace too small): one wave per triple, VALU f32 dot products.
// ---------------------------------------------------------------------------
__global__ __launch_bounds__(256) void direct_kernel(const float* __restrict__ E,
                                                     const float* __restrict__ W,
                                                     const float* __restrict__ b,
                                                     const int* __restrict__ tri,
                                                     float* __restrict__ out) {
    int wave = (int)((blockIdx.x * (size_t)blockDim.x + threadIdx.x) >> 5);
    int lane = threadIdx.x & 31;
    if (wave >= N_TRIPLES) return;
    int s = tri[wave];
    int r = tri[N_TRIPLES + wave];
    int o = tri[2 * N_TRIPLES + wave];
    const float* sv = E + (size_t)s * N_DIM;
    const float* ov = E + (size_t)o * N_DIM;
    const float* wr = W + (size_t)r * (2 * N_DIM);
    float acc = 0.f;
#pragma unroll 4
    for (int k = lane; k < N_DIM; k += 32) {
        acc = fmaf(sv[k], wr[k], acc);
        acc = fmaf(ov[k], wr[N_DIM + k], acc);
    }
#pragma unroll
    for (int off = 16; off > 0; off >>= 1)
        acc += __shfl_down(acc, off, 32);
    if (lane == 0) out[wave] = acc + b[r];
}

extern "C" void kernel_launch(void* const* d_in, const int* in_sizes, int n_in,
                              void* d_out, int out_size, void* d_ws, size_t ws_size,
                              hipStream_t stream) {
    const float* E   = (const float*)d_in[0];   // (100000, 512)  f32
    const float* W   = (const float*)d_in[1];   // (64, 1024)     f32
    const float* b   = (const float*)d_in[2];   // (64,)          f32
    const int*   tri = (const int*)d_in[3];     // (3, 200000)    i32
    float*       out = (float*)d_out;           // (200000,)      f32

    const size_t needed = (size_t)N_NODES * NCOLS * sizeof(float);  // 51.2 MB
    if (ws_size >= needed) {
        float* P = (float*)d_ws;
        proj_kernel<<<N_NODES / TILE_M, 256, 0, stream>>>(E, W, P);
        score_kernel<<<(N_TRIPLES + 255) / 256, 256, 0, stream>>>(tri, P, b, out);
    } else {
        const size_t threads = (size_t)N_TRIPLES * 32;
        direct_kernel<<<(int)((threads + 255) / 256), 256, 0, stream>>>(E, W, b, tri, out);
    }
}